// TTMultiHeadAttention_223338299699
// MI455X (gfx1250) — compile-verified
//
#include <hip/hip_runtime.h>

typedef __attribute__((ext_vector_type(2))) float v2f;
typedef __attribute__((ext_vector_type(8))) float v8f;

#define EMB   1024
#define NTOK  4096
#define LQ    2048
#define HEADS 16
#define HD    64
#define S_STRIDE 2056  // 2048 + 8 floats pad to break LDS bank alignment

static __device__ __forceinline__ v8f wmma_f32(v2f a, v2f b, v8f c) {
  // V_WMMA_F32_16X16X4_F32 : D(16x16,f32) = A(16x4,f32) * B(4x16,f32) + C
  return __builtin_amdgcn_wmma_f32_16x16x4_f32(false, a, false, b, (short)0, c,
                                               false, false);
}

// ---------------------------------------------------------------------------
// Materialize the dense TT matrix, stored TRANSPOSED: Wt[out*1024 + in].
//   W[in,out] = sum_{r1,r2} G0[0,i1,o1,r1] * G1[r1,i2,o2,r2] * G2[r2,i3,o3,0]
//   in  = i1*128 + i2*8 + i3   (i1<8, i2<16, i3<8)
//   out = o1*128 + o2*8 + o3
// ---------------------------------------------------------------------------
__global__ void __launch_bounds__(256)
build_w(const float* __restrict__ g0, const float* __restrict__ g1,
        const float* __restrict__ g2, float* __restrict__ Wt) {
  const int idx = blockIdx.x * 256 + threadIdx.x;  // = out*1024 + in
  const int in  = idx & 1023;
  const int out = idx >> 10;
  const int i1 = in >> 7, i2 = (in >> 3) & 15, i3 = in & 7;
  const int o1 = out >> 7, o2 = (out >> 3) & 15, o3 = out & 7;

  float c2v[16];
#pragma unroll
  for (int r2 = 0; r2 < 16; ++r2) c2v[r2] = g2[(r2 * 8 + i3) * 8 + o3];

  float acc = 0.f;
#pragma unroll 4
  for (int r1 = 0; r1 < 16; ++r1) {
    const float a = g0[(i1 * 8 + o1) * 16 + r1];
    const float* p = g1 + ((r1 * 16 + i2) * 16 + o2) * 16;
    float t = 0.f;
#pragma unroll
    for (int r2 = 0; r2 < 16; ++r2) t = fmaf(p[r2], c2v[r2], t);
    acc = fmaf(a, t, acc);
  }
  Wt[idx] = acc;
}

// ---------------------------------------------------------------------------
// Dense GEMM with fused bias: Y[m,n] = sum_k X[m,k] * Wt[n,k] + bias[n]
// One wave -> 32(M) x 64(N) tile: two A row-tiles share four B tiles
// (0.75 loads per WMMA). K = 1024 in steps of 4 (f32 WMMA, f32 accumulate).
// transposeOut: store Y^T as [n*NTOK + m] (used for V).
// ---------------------------------------------------------------------------
__global__ void __launch_bounds__(128)
tt_gemm(const float* __restrict__ X, const float* __restrict__ Wt,
        const float* __restrict__ bias, float* __restrict__ Y,
        int transposeOut) {
  const int tid  = threadIdx.x;
  const int wave = tid >> 5;
  const int lane = tid & 31;
  const int half = lane >> 4;
  const int ln   = lane & 15;

  const int wid = blockIdx.x * 4 + wave;  // 0..2047
  const int m0  = (wid >> 4) * 32;        // token tile (32 rows)
  const int n0  = (wid & 15) * 64;        // output-col group

  const float* xa0 = X  + (size_t)(m0 + ln) * EMB + 2 * half;       // A tile 0
  const float* xa1 = X  + (size_t)(m0 + 16 + ln) * EMB + 2 * half;  // A tile 1
  const float* wb  = Wt + (size_t)(n0 + ln) * EMB + 2 * half;       // B tiles

  v8f acc[8];
#pragma unroll
  for (int t = 0; t < 8; ++t) acc[t] = (v8f){};

  for (int kk = 0; kk < EMB; kk += 64) {
    // keep the 23.3 TB/s path fed: speculative prefetch one chunk ahead
    __builtin_prefetch(xa0 + kk + 64, 0, 0);
    __builtin_prefetch(xa1 + kk + 64, 0, 0);
#pragma unroll
    for (int t = 0; t < 4; ++t)
      __builtin_prefetch(wb + (size_t)t * 16 * EMB + kk + 64, 0, 0);

#pragma unroll
    for (int k0 = kk; k0 < kk + 64; k0 += 4) {
      const v2f a0 = *(const v2f*)(xa0 + k0);
      const v2f a1 = *(const v2f*)(xa1 + k0);
      v2f b[4];
#pragma unroll
      for (int t = 0; t < 4; ++t)
        b[t] = *(const v2f*)(wb + (size_t)t * 16 * EMB + k0);
#pragma unroll
      for (int t = 0; t < 4; ++t) {
        acc[t]     = wmma_f32(a0, b[t], acc[t]);
        acc[4 + t] = wmma_f32(a1, b[t], acc[4 + t]);
      }
    }
  }

#pragma unroll
  for (int mi = 0; mi < 2; ++mi) {
#pragma unroll
    for (int v = 0; v < 8; ++v) {
      const int m = m0 + 16 * mi + v + 8 * half;
#pragma unroll
      for (int t = 0; t < 4; ++t) {
        const int n = n0 + 16 * t + ln;
        const float y = acc[4 * mi + t][v] + bias[n];
        if (transposeOut)
          Y[(size_t)n * NTOK + m] = y;
        else
          Y[(size_t)m * EMB + n] = y;
      }
    }
  }
}

// ---------------------------------------------------------------------------
// Attention: one workgroup = (batch b, head h, 16-query tile).
// 128 threads (4 waves). Score strip S[16][2048] lives in LDS (128+KB of the
// 320KB/WGP). Phase1: S = (Q K^T)/8 via WMMA. Phase2: row softmax (wave32
// shuffle reduce). Phase3: ctx = S * V via WMMA (A from LDS, B from V^T).
// ---------------------------------------------------------------------------
__global__ void __launch_bounds__(128)
attn(const float* __restrict__ Qb, const float* __restrict__ Kb,
     const float* __restrict__ Vt, float* __restrict__ Cb) {
  extern __shared__ float S[];  // [16][S_STRIDE]
  const int tid  = threadIdx.x;
  const int wave = tid >> 5;
  const int lane = tid & 31;
  const int half = lane >> 4;
  const int ln   = lane & 15;

  const int qt = blockIdx.x & 127;
  const int h  = (blockIdx.x >> 7) & 15;
  const int b  = blockIdx.x >> 11;
  const int tb = b * LQ;
  const int q0 = tb + qt * 16;

  // ---- Phase 1: scores ----
  const float* qa = Qb + (size_t)(q0 + ln) * EMB + h * HD + 2 * half;
  for (int kt = wave; kt < LQ / 16; kt += 4) {  // 32 tiles per wave, uniform
    const float* kb = Kb + (size_t)(tb + kt * 16 + ln) * EMB + h * HD + 2 * half;
    v8f acc = {};
#pragma unroll
    for (int k0 = 0; k0 < HD; k0 += 4) {
      const v2f a  = *(const v2f*)(qa + k0);
      const v2f bv = *(const v2f*)(kb + k0);
      acc = wmma_f32(a, bv, acc);
    }
#pragma unroll
    for (int v = 0; v < 8; ++v)
      S[(v + 8 * half) * S_STRIDE + kt * 16 + ln] = acc[v] * 0.125f;
  }
  __syncthreads();

  // ---- Phase 2: softmax over each of the 16 rows (4 rows per wave) ----
  for (int r = wave * 4; r < wave * 4 + 4; ++r) {
    float* row = S + r * S_STRIDE;
    float mx = -3.4e38f;
    for (int i = lane; i < LQ; i += 32) mx = fmaxf(mx, row[i]);
#pragma unroll
    for (int off = 16; off > 0; off >>= 1) mx = fmaxf(mx, __shfl_xor(mx, off, 32));
    float sum = 0.f;
    for (int i = lane; i < LQ; i += 32) {
      const float e = __expf(row[i] - mx);
      row[i] = e;
      sum += e;
    }
#pragma unroll
    for (int off = 16; off > 0; off >>= 1) sum += __shfl_xor(sum, off, 32);
    const float inv = 1.0f / sum;
    for (int i = lane; i < LQ; i += 32) row[i] *= inv;
  }
  __syncthreads();

  // ---- Phase 3: ctx(16 x 64) = S(16 x 2048) * V(2048 x 64) ----
  const int n0 = wave * 16;  // head-dim column tile for this wave
  const float* vb = Vt + (size_t)(h * HD + n0 + ln) * NTOK + tb + 2 * half;
  v8f acc = {};
#pragma unroll 4
  for (int k0 = 0; k0 < LQ; k0 += 4) {
    const v2f a  = *(const v2f*)(S + ln * S_STRIDE + k0 + 2 * half);
    const v2f bv = *(const v2f*)(vb + k0);
    acc = wmma_f32(a, bv, acc);
  }
#pragma unroll
  for (int v = 0; v < 8; ++v)
    Cb[(size_t)(q0 + v + 8 * half) * EMB + h * HD + n0 + ln] = acc[v];
}

// ---------------------------------------------------------------------------
extern "C" void kernel_launch(void* const* d_in, const int* in_sizes, int n_in,
                              void* d_out, int out_size, void* d_ws,
                              size_t ws_size, hipStream_t stream) {
  (void)in_sizes; (void)n_in; (void)out_size; (void)ws_size;
  const float* query = (const float*)d_in[0];

  // Workspace layout (floats): 4x Wt (1M each) | Q | K | V^T | CTX (4M each)
  float* W  = (float*)d_ws;
  float* Qb = W  + 4 * EMB * EMB;
  float* Kb = Qb + (size_t)NTOK * EMB;
  float* Vt = Kb + (size_t)NTOK * EMB;
  float* Cb = Vt + (size_t)NTOK * EMB;

  // Inputs: [0]=query, then per layer l in {wq,wk,wv,wo}: c0=1+4l c1=2+4l c2=3+4l b=4+4l
  for (int l = 0; l < 4; ++l) {
    build_w<<<4096, 256, 0, stream>>>(
        (const float*)d_in[1 + 4 * l], (const float*)d_in[2 + 4 * l],
        (const float*)d_in[3 + 4 * l], W + (size_t)l * EMB * EMB);
  }

  tt_gemm<<<512, 128, 0, stream>>>(query, W + 0 * EMB * EMB,
                                   (const float*)d_in[4], Qb, 0);
  tt_gemm<<<512, 128, 0, stream>>>(query, W + 1 * EMB * EMB,
                                   (const float*)d_in[8], Kb, 0);
  tt_gemm<<<512, 128, 0, stream>>>(query, W + 2 * EMB * EMB,
                                   (const float*)d_in[12], Vt, /*transpose=*/1);

  attn<<<4096, 128, 16 * S_STRIDE * sizeof(float), stream>>>(Qb, Kb, Vt, Cb);

  tt_gemm<<<512, 128, 0, stream>>>(Cb, W + 3 * EMB * EMB,
                                   (const float*)d_in[16], (float*)d_out, 0);
}